// ConvRelu_28664611733895
// MI455X (gfx1250) — compile-verified
//
#include <hip/hip_runtime.h>

// ---------------------------------------------------------------------------
// Lattice convolution (gather + GEMM + bias + ReLU) for MI455X / gfx1250.
//   rows[i] = concat(lv[i], lv[nbr[i][0..7]])          (576 fp32)
//   out[i]  = relu(rows[i] @ W + b)                    (64 fp32)
//
// Roofline: 37 GFLOP vs ~288 MB HBM (192MB L2 holds the whole 128MB lv table,
// so the 1.15GB gather is L2 traffic). Kernel is data-movement bound; we use
// exact-fp32 V_WMMA_F32_16X16X4_F32 since the matrix pipe is not the limit.
//
// Structure per 256-thread block (8 wave32 waves, persistent over chunks):
//  - transposed W staged once in LDS (64 x 580 floats, padded stride:
//    580 % 64banks = 4 -> conflict-free per-lane b64 fragment loads)
//  - per chunk: gather 64 rows x 576 floats into LDS with coalesced
//    float4 loads (16 lanes = one contiguous 256B lv row)
//  - 8 waves = 4 row-stripes x 2 col-halves; each wave computes 16x32
//    (two 16x16 tiles) sharing the A fragment -> 1.5 ds_load_b64 per WMMA
// ---------------------------------------------------------------------------

typedef __attribute__((ext_vector_type(2))) float v2f;
typedef __attribute__((ext_vector_type(4))) float f4v;
typedef __attribute__((ext_vector_type(8))) float v8f;

#define N_VERT   500000
#define IN_CH    64
#define NF       64
#define FE       9
#define KDIM     (FE * IN_CH)              // 576
#define KF4      (KDIM / 4)                // 144 float4 per row
#define STR      580                       // padded LDS row stride (floats)
#define ROWS_PB  64                        // vertices per chunk
#define NCHUNK   ((N_VERT + ROWS_PB - 1) / ROWS_PB)   // 7813 (last chunk half)
#define THREADS  256                       // 8 wave32 waves

__global__ __launch_bounds__(THREADS)
void lattice_conv_wmma(const float* __restrict__ lv,
                       const long long* __restrict__ nbr,
                       const float* __restrict__ W,
                       const float* __restrict__ bias,
                       float* __restrict__ out)
{
    extern __shared__ float smem[];
    float* wt_s  = smem;                   // [NF][STR]      W^T: wt_s[n*STR + k]
    float* row_s = smem + NF * STR;        // [ROWS_PB][STR] gathered rows

    const int tid  = threadIdx.x;
    const int lane = tid & 31;
    const int wv   = tid >> 5;             // wave id 0..7
    const int half = lane >> 4;            // hi/lo half of wave
    const int m    = lane & 15;            // lane within half

    // ---- stage W (576x64 row-major) transposed into LDS: once per block ----
    const f4v* W4 = reinterpret_cast<const f4v*>(W);
    for (int idx = tid; idx < KDIM * (NF / 4); idx += THREADS) {
        const int k  = idx >> 4;           // 0..575
        const int n  = (idx & 15) * 4;     // filter group
        const f4v w  = W4[idx];            // coalesced 16B read
        wt_s[(n + 0) * STR + k] = w.x;
        wt_s[(n + 1) * STR + k] = w.y;
        wt_s[(n + 2) * STR + k] = w.z;
        wt_s[(n + 3) * STR + k] = w.w;
    }

    const int   stripe = wv >> 1;          // 0..3 : 16-row stripe within chunk
    const int   col0   = (wv & 1) * 32;    // 0 / 32: 32-filter half
    const float bv0    = bias[col0 + m];        // every element in a lane: N = col + m
    const float bv1    = bias[col0 + 16 + m];

    const f4v* lv4 = reinterpret_cast<const f4v*>(lv);

    for (int chunk = blockIdx.x; chunk < NCHUNK; chunk += gridDim.x) {
        __syncthreads();                   // row_s free (wt_s ready on 1st iter)

        // ---- gather 64 rows x 576 floats into LDS ----
        const int base = chunk * ROWS_PB;
        for (int idx = tid; idx < ROWS_PB * KF4; idx += THREADS) {
            const int r   = idx / KF4;
            const int f4  = idx - r * KF4;
            const int seg = f4 >> 4;       // 0 = center, 1..8 = neighbors
            int gr = base + r;
            if (gr >= N_VERT) gr = N_VERT - 1;     // tail clamp: loads stay valid
            const long long v = (seg == 0)
                ? (long long)gr
                : nbr[(size_t)gr * (FE - 1) + (seg - 1)];
            const f4v val = lv4[(size_t)v * (IN_CH / 4) + (f4 & 15)];
            *reinterpret_cast<f4v*>(&row_s[r * STR + f4 * 4]) = val;
        }
        __syncthreads();

        // ---- 16x32 per wave: 2x144 v_wmma_f32_16x16x4_f32, shared A ----
        // A layout (16x4 fp32): lanes 0-15 M=lane K={k0,k0+1};
        //                       lanes 16-31 M=lane-16 K={k0+2,k0+3}
        // -> per-lane b64 at rows[m][k0 + 2*half]; B mirrored from W^T.
        const float* arow = &row_s[(stripe * 16 + m) * STR + 2 * half];
        const float* b0p  = &wt_s [(col0 + m)        * STR + 2 * half];
        const float* b1p  = &wt_s [(col0 + 16 + m)   * STR + 2 * half];

        v8f acc0 = { bv0, bv0, bv0, bv0, bv0, bv0, bv0, bv0 };
        v8f acc1 = { bv1, bv1, bv1, bv1, bv1, bv1, bv1, bv1 };
        #pragma unroll 8
        for (int k0 = 0; k0 < KDIM; k0 += 4) {
            const v2f a  = *reinterpret_cast<const v2f*>(arow + k0);
            const v2f b0 = *reinterpret_cast<const v2f*>(b0p  + k0);
            const v2f b1 = *reinterpret_cast<const v2f*>(b1p  + k0);
            acc0 = __builtin_amdgcn_wmma_f32_16x16x4_f32(
                false, a, false, b0, (short)0, acc0, false, false);
            acc1 = __builtin_amdgcn_wmma_f32_16x16x4_f32(
                false, a, false, b1, (short)0, acc1, false, false);
        }

        // ---- ReLU + store (C/D layout: VGPR vi -> M = vi + 8*half, N = m) ----
        const int grow0 = base + stripe * 16 + 8 * half;
        #pragma unroll
        for (int vi = 0; vi < 8; ++vi) {
            const int grow = grow0 + vi;
            if (grow < N_VERT) {
                float r0 = acc0[vi]; r0 = r0 > 0.0f ? r0 : 0.0f;
                float r1 = acc1[vi]; r1 = r1 > 0.0f ? r1 : 0.0f;
                out[(size_t)grow * NF + (col0 + m)]      = r0;
                out[(size_t)grow * NF + (col0 + 16 + m)] = r1;
            }
        }
    }
}

extern "C" void kernel_launch(void* const* d_in, const int* in_sizes, int n_in,
                              void* d_out, int out_size, void* d_ws, size_t ws_size,
                              hipStream_t stream)
{
    const float*     lv  = (const float*)d_in[0];
    const long long* nbr = (const long long*)d_in[1];   // jnp.int64
    const float*     W   = (const float*)d_in[2];
    const float*     b   = (const float*)d_in[3];
    float*           out = (float*)d_out;

    const size_t shmem = (size_t)(NF * STR + ROWS_PB * STR) * sizeof(float); // 296,960 B < 320 KB/WGP
    (void)hipFuncSetAttribute(reinterpret_cast<const void*>(lattice_conv_wmma),
                              hipFuncAttributeMaxDynamicSharedMemorySize, (int)shmem);

    const int grid = 2048;  // persistent blocks: W staged once per block
    lattice_conv_wmma<<<grid, THREADS, shmem, stream>>>(lv, nbr, W, b, out);
}